// MY_RWKV_RNN_31164282700009
// MI455X (gfx1250) — compile-verified
//
#include <hip/hip_runtime.h>
#include <math.h>

// ---- model dims ----
#define DD 2048
#define FF 7168
#define VV 32000
#define NH 32
#define HS 64
#define NL 8
#define SROWS 66      // 2 + HEAD_SIZE state rows per layer
#define SEG8 8        // split-K segments for D-wide GEMVs (Wo, ffn_Wr, ffn_Wv)
#define SEG4 4        // split-K segments for rkvg batch and ffn_Wk

typedef float v2f __attribute__((ext_vector_type(2)));
typedef float v8f __attribute__((ext_vector_type(8)));

// ============ embed + ln0 ============
__global__ __launch_bounds__(256) void k_embed_ln(const int* __restrict__ tok,
                                                  const float* __restrict__ emb,
                                                  const float* __restrict__ w,
                                                  const float* __restrict__ b,
                                                  float* __restrict__ out) {
  __shared__ float sb[256];
  const float* x = emb + (size_t)tok[0] * DD;
  int tid = threadIdx.x;
  float s = 0.f;
  for (int j = tid; j < DD; j += 256) s += x[j];
  sb[tid] = s; __syncthreads();
  #pragma unroll
  for (int t = 128; t > 0; t >>= 1) { if (tid < t) sb[tid] += sb[tid + t]; __syncthreads(); }
  float mean = sb[0] * (1.0f / DD); __syncthreads();
  s = 0.f;
  for (int j = tid; j < DD; j += 256) { float d = x[j] - mean; s += d * d; }
  sb[tid] = s; __syncthreads();
  #pragma unroll
  for (int t = 128; t > 0; t >>= 1) { if (tid < t) sb[tid] += sb[tid + t]; __syncthreads(); }
  float rstd = rsqrtf(sb[0] * (1.0f / DD) + 1e-5f);
  for (int j = tid; j < DD; j += 256) out[j] = (x[j] - mean) * rstd * w[j] + b[j];
}

// ============ time-mix prep ============
// x += sigmoid(sum Wr-parts)*sum Wv-parts (prev layer ffn, if present); then
// xn = LN(x); sx = st_in - xn; st_out = xn; xxx = xn + sx*maa_x
__global__ __launch_bounds__(256) void k_time_prep(float* __restrict__ X,
                                                   const float* __restrict__ vparts,
                                                   const float* __restrict__ rparts,
                                                   const float* __restrict__ w,
                                                   const float* __restrict__ b,
                                                   const float* __restrict__ stin,
                                                   float* __restrict__ stout,
                                                   const float* __restrict__ maax,
                                                   float* __restrict__ xn,
                                                   float* __restrict__ sx,
                                                   float* __restrict__ xxx) {
  __shared__ float xb[DD];
  __shared__ float sb[256];
  int tid = threadIdx.x;
  for (int j = tid; j < DD; j += 256) {
    float xj = X[j];
    if (vparts) {
      float sv = 0.f, sr = 0.f;
      #pragma unroll
      for (int s = 0; s < SEG8; ++s) { sv += vparts[s * DD + j]; sr += rparts[s * DD + j]; }
      xj += sv / (1.f + expf(-sr));
      X[j] = xj;
    }
    xb[j] = xj;
  }
  __syncthreads();
  float s = 0.f;
  for (int j = tid; j < DD; j += 256) s += xb[j];
  sb[tid] = s; __syncthreads();
  #pragma unroll
  for (int t = 128; t > 0; t >>= 1) { if (tid < t) sb[tid] += sb[tid + t]; __syncthreads(); }
  float mean = sb[0] * (1.0f / DD); __syncthreads();
  s = 0.f;
  for (int j = tid; j < DD; j += 256) { float d = xb[j] - mean; s += d * d; }
  sb[tid] = s; __syncthreads();
  #pragma unroll
  for (int t = 128; t > 0; t >>= 1) { if (tid < t) sb[tid] += sb[tid + t]; __syncthreads(); }
  float rstd = rsqrtf(sb[0] * (1.0f / DD) + 1e-5f);
  for (int j = tid; j < DD; j += 256) {
    float n = (xb[j] - mean) * rstd * w[j] + b[j];
    float s0 = stin[j] - n;
    xn[j] = n; sx[j] = s0; stout[j] = n;
    xxx[j] = n + s0 * maax[j];
  }
}

// ============ channel-mix prep (folds Wo split-K partials into residual) ============
__global__ __launch_bounds__(256) void k_chan_prep(float* __restrict__ X,
                                                   const float* __restrict__ woparts,
                                                   const float* __restrict__ w,
                                                   const float* __restrict__ b,
                                                   const float* __restrict__ stin,
                                                   float* __restrict__ stout,
                                                   const float* __restrict__ maak,
                                                   const float* __restrict__ maar,
                                                   float* __restrict__ xk,
                                                   float* __restrict__ xr) {
  __shared__ float xb[DD];
  __shared__ float sb[256];
  int tid = threadIdx.x;
  for (int j = tid; j < DD; j += 256) {
    float acc = X[j];
    #pragma unroll
    for (int s = 0; s < SEG8; ++s) acc += woparts[s * DD + j];
    X[j] = acc; xb[j] = acc;
  }
  __syncthreads();
  float s = 0.f;
  for (int j = tid; j < DD; j += 256) s += xb[j];
  sb[tid] = s; __syncthreads();
  #pragma unroll
  for (int t = 128; t > 0; t >>= 1) { if (tid < t) sb[tid] += sb[tid + t]; __syncthreads(); }
  float mean = sb[0] * (1.0f / DD); __syncthreads();
  s = 0.f;
  for (int j = tid; j < DD; j += 256) { float d = xb[j] - mean; s += d * d; }
  sb[tid] = s; __syncthreads();
  #pragma unroll
  for (int t = 128; t > 0; t >>= 1) { if (tid < t) sb[tid] += sb[tid + t]; __syncthreads(); }
  float rstd = rsqrtf(sb[0] * (1.0f / DD) + 1e-5f);
  for (int j = tid; j < DD; j += 256) {
    float n = (xb[j] - mean) * rstd * w[j] + b[j];
    float s0 = stin[j] - n;
    stout[j] = n;
    xk[j] = n + s0 * maak[j];
    xr[j] = n + s0 * maar[j];
  }
}

// ============ final LN (folds last ffn partials) ============
__global__ __launch_bounds__(256) void k_final_ln(const float* __restrict__ X,
                                                  const float* __restrict__ vparts,
                                                  const float* __restrict__ rparts,
                                                  const float* __restrict__ w,
                                                  const float* __restrict__ b,
                                                  float* __restrict__ out) {
  __shared__ float xb[DD];
  __shared__ float sb[256];
  int tid = threadIdx.x;
  for (int j = tid; j < DD; j += 256) {
    float sv = 0.f, sr = 0.f;
    #pragma unroll
    for (int s = 0; s < SEG8; ++s) { sv += vparts[s * DD + j]; sr += rparts[s * DD + j]; }
    xb[j] = X[j] + sv / (1.f + expf(-sr));
  }
  __syncthreads();
  float s = 0.f;
  for (int j = tid; j < DD; j += 256) s += xb[j];
  sb[tid] = s; __syncthreads();
  #pragma unroll
  for (int t = 128; t > 0; t >>= 1) { if (tid < t) sb[tid] += sb[tid + t]; __syncthreads(); }
  float mean = sb[0] * (1.0f / DD); __syncthreads();
  s = 0.f;
  for (int j = tid; j < DD; j += 256) { float d = xb[j] - mean; s += d * d; }
  sb[tid] = s; __syncthreads();
  #pragma unroll
  for (int t = 128; t > 0; t >>= 1) { if (tid < t) sb[tid] += sb[tid + t]; __syncthreads(); }
  float rstd = rsqrtf(sb[0] * (1.0f / DD) + 1e-5f);
  for (int j = tid; j < DD; j += 256) out[j] = (xb[j] - mean) * rstd * w[j] + b[j];
}

// ============ fused token-shift mixer (one 1024-thread block) ============
// t160 = tanh(xxx @ w1); xw..xg = xn + sx*(maa_f + t160[f]@w2[f]);
// d64 = tanh(xw @ dw1); wdec = exp(-exp(decay + d64 @ dw2))
__global__ __launch_bounds__(1024) void k_mixer(const float* __restrict__ xxx,
                                                const float* __restrict__ xn,
                                                const float* __restrict__ sx,
                                                const float* __restrict__ w1,
                                                const float* __restrict__ w2,
                                                const float* __restrict__ dw1,
                                                const float* __restrict__ dw2,
                                                const float* __restrict__ decay,
                                                const float* __restrict__ maaw,
                                                const float* __restrict__ maak,
                                                const float* __restrict__ maav,
                                                const float* __restrict__ maar,
                                                const float* __restrict__ maag,
                                                float* __restrict__ xm,   // k,v,r,g (4*DD)
                                                float* __restrict__ wdec) {
  __shared__ float xw[DD];      // reused: phase A holds xxx, phase B+ holds xw
  __shared__ float ts[160];
  __shared__ float pA[640];
  __shared__ float ds64[64];
  __shared__ float pC[1024];
  int tid = threadIdx.x;

  // stage xxx into LDS
  xw[tid] = xxx[tid]; xw[tid + 1024] = xxx[tid + 1024];
  __syncthreads();

  // phase A: t160 (4 d-chunks x 160 outputs)
  if (tid < 640) {
    int c = tid / 160, t = tid - c * 160;
    float acc = 0.f;
    for (int d = c * 512; d < c * 512 + 512; ++d)
      acc = fmaf(xw[d], w1[(size_t)d * 160 + t], acc);
    pA[tid] = acc;
  }
  __syncthreads();
  if (tid < 160) ts[tid] = tanhf(pA[tid] + pA[160 + tid] + pA[320 + tid] + pA[480 + tid]);
  __syncthreads();

  // phase B: 5-way mixed shifts (2 channels per thread)
  const float* maas[5] = {maaw, maak, maav, maar, maag};
  #pragma unroll
  for (int rep = 0; rep < 2; ++rep) {
    int j = tid + rep * 1024;
    float xnj = xn[j], sxj = sx[j];
    #pragma unroll
    for (int f = 0; f < 5; ++f) {
      float m = 0.f;
      #pragma unroll 4
      for (int t = 0; t < 32; ++t)
        m = fmaf(ts[f * 32 + t], w2[(size_t)(f * 32 + t) * DD + j], m);
      float val = xnj + sxj * (maas[f][j] + m);
      if (f == 0) xw[j] = val;                 // xw kept in LDS
      else        xm[(f - 1) * DD + j] = val;  // xk,xv,xr,xg to global
    }
  }
  __syncthreads();

  // phase C: d64 (16 d-chunks x 64 outputs)
  {
    int c = tid >> 6, t = tid & 63;
    float acc = 0.f;
    for (int d = c * 128; d < c * 128 + 128; ++d)
      acc = fmaf(xw[d], dw1[(size_t)d * 64 + t], acc);
    pC[tid] = acc;
  }
  __syncthreads();
  if (tid < 64) {
    float acc = 0.f;
    #pragma unroll
    for (int c = 0; c < 16; ++c) acc += pC[c * 64 + tid];
    ds64[tid] = tanhf(acc);
  }
  __syncthreads();

  // phase D: decay vector
  #pragma unroll
  for (int rep = 0; rep < 2; ++rep) {
    int j = tid + rep * 1024;
    float acc = decay[j];
    #pragma unroll 4
    for (int t = 0; t < 64; ++t) acc = fmaf(ds64[t], dw2[(size_t)t * DD + j], acc);
    wdec[j] = expf(-expf(acc));
  }
}

// ============ split-K WMMA GEMV: writes raw per-segment partials ============
// One wave owns 16 rows; A = 16x4 fp32 weight tile (ISA 7.12.2 layout: lanes
// 0-15 K0/K1, lanes 16-31 K2/K3), B = x broadcast over all 16 columns from LDS.
// Optional fused staging: xs = relu(sum of xsegs partials)^2 (ffn kk path).
struct GemvP { const float* W[4]; const float* x[4]; float* part[4]; };

__global__ __launch_bounds__(256) void k_gemv_part(GemvP gp, int K, int M, int bpm,
                                                   int segs, const float* xparts,
                                                   int xsegs, int xlen) {
  int Kseg = K / segs;
  int idx = blockIdx.x;
  int sub = idx / (bpm * segs);
  int rem = idx - sub * (bpm * segs);
  int blk = rem / segs;
  int seg = rem - blk * segs;

  const float* W = gp.W[sub];
  float* part = gp.part[sub];

  extern __shared__ float xs[];
  for (int j = threadIdx.x; j < Kseg; j += 256) {
    int gj = seg * Kseg + j;
    float a;
    if (xparts) {
      a = 0.f;
      for (int t = 0; t < xsegs; ++t) a += xparts[(size_t)t * xlen + gj];
      a = fmaxf(a, 0.f); a = a * a;            // relu^2 (ffn k path)
    } else {
      a = gp.x[sub][gj];
    }
    xs[j] = a;
  }
  __syncthreads();

  int wave = threadIdx.x >> 5;
  int lane = threadIdx.x & 31;
  int m0   = blk * 128 + wave * 16;
  int rowA = m0 + (lane & 15);
  int koff = (lane >> 4) << 1;                  // 0 for lanes 0-15, 2 for 16-31
  const float* wp = W + (size_t)rowA * K + (size_t)seg * Kseg + koff;

  v8f c = {};
  for (int kc = 0; kc < Kseg; kc += 4) {
    v2f a = *(const v2f*)(wp + kc);
    v2f b;
    b.x = xs[kc + koff];
    b.y = xs[kc + koff + 1];
    c = __builtin_amdgcn_wmma_f32_16x16x4_f32(false, a, false, b,
                                              (short)0, c, false, false);
  }

  if ((lane & 15) == 0) {                       // lanes 0 and 16: M 0-7 / 8-15
    int mbase = m0 + ((lane >> 4) << 3);
    #pragma unroll
    for (int t = 0; t < 8; ++t) part[(size_t)seg * M + mbase + t] = c[t];
  }
}

// ============ per-head WKV + groupnorm + SiLU gate (fused) ============
// Sums the rkvg split-K partials while loading. state' written directly to out.
__global__ __launch_bounds__(256) void k_attn_gn(const float* __restrict__ P,   // rkvg partials
                                                 const float* __restrict__ wdec,
                                                 const float* __restrict__ ulay,
                                                 const float* __restrict__ stin,
                                                 float* __restrict__ stout,
                                                 const float* __restrict__ lnxw,
                                                 const float* __restrict__ lnxb,
                                                 float* __restrict__ z) {
  __shared__ float rs[HS], ks[HS], vs[HS], wv[HS], us[HS];
  __shared__ float red[256];
  __shared__ float rw[HS], sb[HS];
  int h = blockIdx.x, tid = threadIdx.x;
  int c = h * HS + (tid & 63);
  float gval = 0.f;
  if (tid < HS) {
    float a = 0.f, bq = 0.f, cv = 0.f, g = 0.f;
    #pragma unroll
    for (int s = 0; s < SEG4; ++s) {
      a  += P[(size_t)(0 * SEG4 + s) * DD + c];
      bq += P[(size_t)(1 * SEG4 + s) * DD + c];
      cv += P[(size_t)(2 * SEG4 + s) * DD + c];
      g  += P[(size_t)(3 * SEG4 + s) * DD + c];
    }
    rs[tid] = a; ks[tid] = bq; vs[tid] = cv; gval = g;
    wv[tid] = wdec[c]; us[tid] = ulay[c];
  }
  __syncthreads();
  int t = tid & 63, sg = tid >> 6;
  const float* pv = stin  + (size_t)h * HS * HS;
  float*       po = stout + (size_t)h * HS * HS;
  float part = 0.f;
  #pragma unroll 4
  for (int s = sg * 16; s < sg * 16 + 16; ++s) {
    float p   = pv[s * HS + t];
    float kvt = ks[s] * vs[t];
    part += rs[s] * fmaf(us[s], kvt, p);
    po[s * HS + t] = fmaf(p, wv[s], kvt);
  }
  red[tid] = part; __syncthreads();
  if (sg == 0) rw[t] = red[t] + red[64 + t] + red[128 + t] + red[192 + t];
  __syncthreads();
  // groupnorm over this head's 64 values
  if (tid < HS) sb[tid] = rw[tid];
  __syncthreads();
  #pragma unroll
  for (int s = 32; s > 0; s >>= 1) { if (tid < s) sb[tid] += sb[tid + s]; __syncthreads(); }
  float mean = sb[0] * (1.f / HS); __syncthreads();
  if (tid < HS) { float d = rw[tid] - mean; sb[tid] = d * d; }
  __syncthreads();
  #pragma unroll
  for (int s = 32; s > 0; s >>= 1) { if (tid < s) sb[tid] += sb[tid + s]; __syncthreads(); }
  float rstd = rsqrtf(sb[0] * (1.f / HS) + 64e-5f);
  if (tid < HS) {
    float y = (rw[tid] - mean) * rstd * lnxw[c] + lnxb[c];
    z[c] = y * gval / (1.f + expf(-gval));      // y * silu(g)
  }
}

// ---------------- workspace layout (floats) ----------------
enum : size_t {
  OX = 0, OXN = OX + DD, OSX = OXN + DD, OXXX = OSX + DD,
  OXM = OXXX + DD,                 // xk,xv,xr,xg : 4*DD
  OWDEC = OXM + 4 * DD,
  OZ    = OWDEC + DD,
  OXK2  = OZ + DD, OXR2 = OXK2 + DD,
  OPRKVG = OXR2 + DD,              // 4 mats * SEG4 * DD = 32768
  OPWO   = OPRKVG + 4 * SEG4 * DD, // SEG8 * DD
  OPR    = OPWO + SEG8 * DD,       // SEG8 * DD
  OPK    = OPR + SEG8 * DD,        // SEG4 * FF
  OPV    = OPK + (size_t)SEG4 * FF // SEG8 * DD
};

extern "C" void kernel_launch(void* const* d_in, const int* in_sizes, int n_in,
                              void* d_out, int out_size, void* d_ws, size_t ws_size,
                              hipStream_t stream) {
  (void)in_sizes; (void)n_in; (void)out_size; (void)ws_size;
  const int*   tok    = (const int*)  d_in[0];
  const float* st_in  = (const float*)d_in[1];
  const float* emb    = (const float*)d_in[2];
  const float* ln0w   = (const float*)d_in[3];
  const float* ln0b   = (const float*)d_in[4];
  const float* ln1w   = (const float*)d_in[5];
  const float* ln1b   = (const float*)d_in[6];
  const float* ln2w   = (const float*)d_in[7];
  const float* ln2b   = (const float*)d_in[8];
  const float* lnow   = (const float*)d_in[9];
  const float* lnob   = (const float*)d_in[10];
  const float* headW  = (const float*)d_in[11];
  const float* maax   = (const float*)d_in[12];
  const float* maaw   = (const float*)d_in[13];
  const float* maak   = (const float*)d_in[14];
  const float* maav   = (const float*)d_in[15];
  const float* maar   = (const float*)d_in[16];
  const float* maag   = (const float*)d_in[17];
  const float* maaw1  = (const float*)d_in[18];
  const float* maaw2  = (const float*)d_in[19];
  const float* decay  = (const float*)d_in[20];
  const float* dw1    = (const float*)d_in[21];
  const float* dw2    = (const float*)d_in[22];
  const float* faaaa  = (const float*)d_in[23];
  const float* aWr    = (const float*)d_in[24];
  const float* aWk    = (const float*)d_in[25];
  const float* aWv    = (const float*)d_in[26];
  const float* aWg    = (const float*)d_in[27];
  const float* aWo    = (const float*)d_in[28];
  const float* lnxw   = (const float*)d_in[29];
  const float* lnxb   = (const float*)d_in[30];
  const float* fmaak  = (const float*)d_in[31];
  const float* fmaar  = (const float*)d_in[32];
  const float* fWr    = (const float*)d_in[33];
  const float* fWk    = (const float*)d_in[34];
  const float* fWv    = (const float*)d_in[35];

  float* ws   = (float*)d_ws;
  float* X    = ws + OX;    float* XN  = ws + OXN;   float* SX  = ws + OSX;
  float* XXX  = ws + OXXX;  float* XM  = ws + OXM;   float* WDE = ws + OWDEC;
  float* Z    = ws + OZ;    float* XK2 = ws + OXK2;  float* XR2 = ws + OXR2;
  float* Prkvg = ws + OPRKVG;
  float* Pwo   = ws + OPWO;
  float* Pr    = ws + OPR;
  float* Pk    = ws + OPK;
  float* Pv    = ws + OPV;

  float* logits = (float*)d_out;
  float* st_out = (float*)d_out + VV;

  k_embed_ln<<<1, 256, 0, stream>>>(tok, emb, ln0w, ln0b, X);

  for (int i = 0; i < NL; ++i) {
    size_t oD = (size_t)i * DD;
    const float* vp = (i == 0) ? nullptr : Pv;   // pending ffn partials from layer i-1
    const float* rp = (i == 0) ? nullptr : Pr;

    // ---- time mixing ----
    k_time_prep<<<1, 256, 0, stream>>>(X, vp, rp, ln1w + oD, ln1b + oD,
                                       st_in  + (size_t)(SROWS * i + 1) * DD,
                                       st_out + (size_t)(SROWS * i + 1) * DD,
                                       maax + oD, XN, SX, XXX);
    k_mixer<<<1, 1024, 0, stream>>>(XXX, XN, SX,
                                    maaw1 + (size_t)i * DD * 160,
                                    maaw2 + (size_t)i * 160 * DD,
                                    dw1 + (size_t)i * DD * 64,
                                    dw2 + (size_t)i * 64 * DD,
                                    decay + oD,
                                    maaw + oD, maak + oD, maav + oD, maar + oD, maag + oD,
                                    XM, WDE);
    GemvP ga{};
    ga.W[0] = aWr + (size_t)i * DD * DD; ga.x[0] = XM + 2 * DD; ga.part[0] = Prkvg;                 // r <- xr
    ga.W[1] = aWk + (size_t)i * DD * DD; ga.x[1] = XM + 0 * DD; ga.part[1] = Prkvg + SEG4 * DD;     // k <- xk
    ga.W[2] = aWv + (size_t)i * DD * DD; ga.x[2] = XM + 1 * DD; ga.part[2] = Prkvg + 2 * SEG4 * DD; // v <- xv
    ga.W[3] = aWg + (size_t)i * DD * DD; ga.x[3] = XM + 3 * DD; ga.part[3] = Prkvg + 3 * SEG4 * DD; // g <- xg
    k_gemv_part<<<4 * 16 * SEG4, 256, (DD / SEG4) * 4, stream>>>(ga, DD, DD, 16, SEG4,
                                                                 nullptr, 0, 0);
    k_attn_gn<<<NH, 256, 0, stream>>>(Prkvg, WDE, faaaa + oD,
                                      st_in  + (size_t)(SROWS * i + 2) * DD,
                                      st_out + (size_t)(SROWS * i + 2) * DD,
                                      lnxw + oD, lnxb + oD, Z);
    GemvP go{};
    go.W[0] = aWo + (size_t)i * DD * DD; go.x[0] = Z; go.part[0] = Pwo;
    k_gemv_part<<<16 * SEG8, 256, (DD / SEG8) * 4, stream>>>(go, DD, DD, 16, SEG8,
                                                             nullptr, 0, 0);

    // ---- channel mixing ----
    k_chan_prep<<<1, 256, 0, stream>>>(X, Pwo, ln2w + oD, ln2b + oD,
                                       st_in  + (size_t)(SROWS * i) * DD,
                                       st_out + (size_t)(SROWS * i) * DD,
                                       fmaak + oD, fmaar + oD, XK2, XR2);
    GemvP gr{};
    gr.W[0] = fWr + (size_t)i * DD * DD; gr.x[0] = XR2; gr.part[0] = Pr;
    k_gemv_part<<<16 * SEG8, 256, (DD / SEG8) * 4, stream>>>(gr, DD, DD, 16, SEG8,
                                                             nullptr, 0, 0);
    GemvP gk{};
    gk.W[0] = fWk + (size_t)i * FF * DD; gk.x[0] = XK2; gk.part[0] = Pk;
    k_gemv_part<<<56 * SEG4, 256, (DD / SEG4) * 4, stream>>>(gk, DD, FF, 56, SEG4,
                                                             nullptr, 0, 0);
    GemvP gv{};
    gv.W[0] = fWv + (size_t)i * DD * FF; gv.x[0] = nullptr; gv.part[0] = Pv;
    k_gemv_part<<<16 * SEG8, 256, (FF / SEG8) * 4, stream>>>(gv, FF, DD, 16, SEG8,
                                                             Pk, SEG4, FF);   // xs = relu(sum Pk)^2
  }

  // x += last ffn; logits = head @ ln(x)  (head is wide: 250 blocks, no split)
  k_final_ln<<<1, 256, 0, stream>>>(X, Pv, Pr, lnow, lnob, XN);
  GemvP gh{};
  gh.W[0] = headW; gh.x[0] = XN; gh.part[0] = logits;
  k_gemv_part<<<250, 256, DD * 4, stream>>>(gh, DD, VV, 250, 1, nullptr, 0, 0);
}